// TdoaLoss_49392123904373
// MI455X (gfx1250) — compile-verified
//
#include <hip/hip_runtime.h>
#include <stdint.h>

// ---------------- problem constants ----------------
#define NBT       32000      // B*T = 64*500
#define BTB       16         // (b,t) pairs per block
#define NBLK      (NBT/BTB)  // 2000 blocks, exact (no tail)
#define NTHREADS  128        // 4 waves (wave32)
#define NP        6          // mic pairs
#define TRK       5          // tracks / slots
#define NCL       13         // logit classes (= NLOGIT) and target classes dim
#define PRED_BT   390        // 13*5*6 floats per (b,t)
#define TGT_BT    325        // 5*5*13 floats per (b,t)
#define ITEMS     (BTB*NP)   // 96 loss items per block
#define NGRP      (ITEMS/16) // 6 WMMA groups of 16 items
#define IGN       (-100)

typedef __attribute__((ext_vector_type(2)))  float        v2f;
typedef __attribute__((ext_vector_type(8)))  float        v8f;
typedef __attribute__((ext_vector_type(4)))  unsigned int v4u;
typedef __attribute__((ext_vector_type(8)))  int          v8i;
typedef __attribute__((ext_vector_type(4)))  int          v4i;

__constant__ int PAIR_A[NP] = {0,0,0,1,1,2};
__constant__ int PAIR_B[NP] = {1,2,3,2,3,3};

// Lexicographic unranking of a permutation of {0..4} (matches itertools.permutations),
// returned as a 25-bit mask with bit (slot*5 + track) set.
__device__ inline uint32_t perm_mask(int idx) {
  int elems[5] = {0,1,2,3,4};
  const int fact[5] = {24,6,2,1,1};
  uint32_t m = 0;
#pragma unroll
  for (int j = 0; j < 5; ++j) {
    int d = idx / fact[j]; idx -= d * fact[j];
    int pj = elems[d];
#pragma unroll
    for (int q = 0; q < 4; ++q)           // shift remaining elements down
      if (q >= d && q < 4 - j) elems[q] = elems[q+1];
    m |= 1u << (j*5 + pj);
  }
  return m;
}

// ---- Tensor Data Mover: 1D contiguous f32 copy global -> LDS (ISA 08, D# groups) ----
__device__ inline void tdm_copy_1d(const float* gsrc, uint32_t lds_byte_off, uint32_t n_elems) {
  uint64_t ga = (uint64_t)(uintptr_t)gsrc;
  v4u g0;
  g0.x = 1u;                                             // count=1, user mode, no gather
  g0.y = lds_byte_off;                                   // lds_addr (bytes)
  g0.z = (uint32_t)ga;                                   // global_addr[31:0]
  g0.w = (uint32_t)((ga >> 32) & 0x01FFFFFFu) | (2u << 30); // addr[56:32] | type=2
  v8i g1;
  g1[0] = (int)(2u << 16);                               // wg_mask=0, data_size=2 (4B)
  g1[1] = (int)((n_elems & 0xFFFFu) << 16);              // tensor_dim0 lo16
  g1[2] = (int)(((n_elems >> 16) & 0xFFFFu) | (1u << 16)); // tensor_dim0 hi | tensor_dim1=1
  g1[3] = (int)((n_elems & 0xFFFFu) << 16);              // tile_dim0 (16b)
  g1[4] = 0;                                             // tile_dim1=0, tile_dim2=0 (unused)
  g1[5] = (int)n_elems;                                  // tensor_dim0_stride lo32
  g1[6] = 0;
  g1[7] = 0;
  v4i z4 = {0,0,0,0};
  v8i z8 = {0,0,0,0,0,0,0,0};
  __builtin_amdgcn_tensor_load_to_lds(g0, g1, z4, z4, z8, 0);
}

__global__ __launch_bounds__(NTHREADS)
void tdoa_main(const float* __restrict__ pred, const float* __restrict__ target,
               const float* __restrict__ mic,
               double* __restrict__ loss_acc,
               unsigned long long* __restrict__ corr_acc,
               unsigned long long* __restrict__ npred_acc)
{
  __shared__ float sPred[BTB * PRED_BT];   // 24960 B
  __shared__ float sTgt [BTB * TGT_BT];    // 20800 B
  __shared__ int   sCls [BTB][TRK][NP];    // target class or IGN
  __shared__ int   sNv  [BTB];             // #valid slots per (b,t)
  __shared__ float sL   [NGRP][16][28];    // vec'd loss matrices, [group][item col][k]
  __shared__ int   sBest[ITEMS];           // best perm index per item
  __shared__ float sVal [ITEMS];           // min cost per item (already /5)
  __shared__ int   sCorrect, sNPred;

  const int t   = threadIdx.x;
  const int bt0 = blockIdx.x * BTB;

  if (t == 0) {
    sCorrect = 0; sNPred = 0;
    // ---- TDM async staging of this block's contiguous slices ----
    tdm_copy_1d(pred   + (size_t)bt0 * PRED_BT, (uint32_t)(uintptr_t)(void*)sPred, BTB*PRED_BT);
    tdm_copy_1d(target + (size_t)bt0 * TGT_BT,  (uint32_t)(uintptr_t)(void*)sTgt,  BTB*TGT_BT);
    __builtin_amdgcn_s_wait_tensorcnt(0);
  }
  __syncthreads();

  // ---------------- phase 1: TDOA target classes (one thread per (b,t)) ----------------
  if (t < BTB) {
    const float* tg = sTgt + t * TGT_BT;
    float mx[4], my[4], mz[4];
#pragma unroll
    for (int m = 0; m < 4; ++m) { mx[m]=mic[m*3+0]; my[m]=mic[m*3+1]; mz[m]=mic[m*3+2]; }
    int count = 0;
    for (int tr = 0; tr < TRK; ++tr) {
      for (int c = 0; c < NCL; ++c) {
        float act = tg[tr*65 + 0*13 + c];
        if (act != 0.0f) {
          if (count < TRK) {
            float dist = tg[tr*65 + 4*13 + c];
            float sx = tg[tr*65 + 1*13 + c] * dist;
            float sy = tg[tr*65 + 2*13 + c] * dist;
            float sz = tg[tr*65 + 3*13 + c] * dist;
            float d[4];
#pragma unroll
            for (int m = 0; m < 4; ++m) {
              float dx = sx - mx[m], dy = sy - my[m], dz = sz - mz[m];
              d[m] = sqrtf(dx*dx + dy*dy + dz*dz);
            }
#pragma unroll
            for (int p = 0; p < NP; ++p) {
              float tdoa = d[PAIR_A[p]] - d[PAIR_B[p]];
              sCls[t][count][p] = (int)rintf((tdoa * 24000.0f) / 343.0f) + 6;
            }
          }
          ++count;
        }
      }
    }
    int nv = count < TRK ? count : TRK;
    sNv[t] = nv;
    for (int s = nv; s < TRK; ++s)
#pragma unroll
      for (int p = 0; p < NP; ++p) sCls[t][s][p] = IGN;
  }
  __syncthreads();

  // ---------------- phase 2: log-softmax + loss matrix per item ----------------
  int nPredLocal = 0;
  if (t < ITEMS) {
    const int bt = t / NP, p = t - bt*NP;
    const int g = t >> 4, col = t & 15;
    const float* pb = sPred + bt * PRED_BT + p;
    float logZ[TRK];
#pragma unroll
    for (int tr = 0; tr < TRK; ++tr) {
      float l[NCL];
#pragma unroll
      for (int cl = 0; cl < NCL; ++cl) l[cl] = pb[cl*30 + tr*6];
      float mx = l[0];
#pragma unroll
      for (int cl = 1; cl < NCL; ++cl) mx = fmaxf(mx, l[cl]);
      float s = 0.0f;
#pragma unroll
      for (int cl = 0; cl < NCL; ++cl) s += expf(l[cl] - mx);
      logZ[tr] = mx + logf(s);
    }
    nPredLocal = sNv[bt];
#pragma unroll
    for (int j = 0; j < TRK; ++j) {
      int c = sCls[bt][j][p];
      if (c == IGN) {
#pragma unroll
        for (int k = 0; k < TRK; ++k) sL[g][col][j*5+k] = 0.0f;
      } else {
#pragma unroll
        for (int k = 0; k < TRK; ++k) sL[g][col][j*5+k] = logZ[k] - pb[c*30 + k*6]; // -logp
      }
    }
    sL[g][col][25] = 0.0f; sL[g][col][26] = 0.0f; sL[g][col][27] = 0.0f;
  }
  __syncthreads();

  // ---------------- phase 3: cost = P(128x28) @ L(28x16) via WMMA f32 16x16x4 ----------------
  {
    const int wv   = t >> 5;
    const int lane = t & 31;
    const int jcol = lane & 15;
    const int hi   = lane >> 4;          // lane half
    uint32_t rmask[8];
#pragma unroll
    for (int m = 0; m < 8; ++m) {
      int r = m*16 + jcol;               // A-matrix row for this lane (M = lane%16)
      rmask[m] = (r < 120) ? perm_mask(r) : 0u;
    }
    const v8f vzero = {0,0,0,0,0,0,0,0};
    for (int g = wv; g < NGRP; g += 4) { // wave-uniform loop: EXEC all-ones for WMMA
      v8f acc[8];
#pragma unroll
      for (int m = 0; m < 8; ++m) acc[m] = vzero;
#pragma unroll
      for (int k0 = 0; k0 < 7; ++k0) {
        const int kb = k0*4 + hi*2;      // same K base for A and B lane layouts
        v2f b; b.x = sL[g][jcol][kb]; b.y = sL[g][jcol][kb+1];
#pragma unroll
        for (int m = 0; m < 8; ++m) {
          v2f a; a.x = (float)((rmask[m] >> kb) & 1u);
                 a.y = (float)((rmask[m] >> (kb+1)) & 1u);
          acc[m] = __builtin_amdgcn_wmma_f32_16x16x4_f32(
              false, a, false, b, (short)0, acc[m], false, false);
        }
      }
      // per-lane argmin over its 64 rows (r = m*16 + v + 8*hi), skip padded rows >=120
      float bv = 3.402823466e38f; int br = 0;
#pragma unroll
      for (int m = 0; m < 8; ++m) {
#pragma unroll
        for (int v = 0; v < 8; ++v) {
          int r = m*16 + v + 8*hi;
          float val = acc[m][v];
          if (r < 120 && (val < bv || (val == bv && r < br))) { bv = val; br = r; }
        }
      }
      // combine lane halves (items live in lanes j and j+16)
      float ov  = __shfl_xor(bv, 16, 32);
      int   orr = __shfl_xor(br, 16, 32);
      if (ov < bv || (ov == bv && orr < br)) { bv = ov; br = orr; }
      if (hi == 0) {
        sBest[g*16 + jcol] = br;
        sVal [g*16 + jcol] = bv / 5.0f;  // cost = mean over 5 slots
      }
    }
  }
  __syncthreads();

  // ---------------- phase 4: accuracy bookkeeping ----------------
  int corrLocal = 0;
  if (t < ITEMS) {
    const int bt = t / NP, p = t - bt*NP;
    const float* pb = sPred + bt * PRED_BT + p;
    uint32_t pm = perm_mask(sBest[t]);
#pragma unroll
    for (int k = 0; k < TRK; ++k) {
      int s = __ffs((pm >> (5*k)) & 31u) - 1;   // best[k] = slot assigned to track k
      int c = sCls[bt][s][p];
      if (c != IGN) {
        float best = pb[0*30 + k*6]; int bi = 0;
#pragma unroll
        for (int cl = 1; cl < NCL; ++cl) {
          float v = pb[cl*30 + k*6];
          if (v > best) { best = v; bi = cl; }
        }
        if (bi == c) ++corrLocal;
      }
    }
  }
  if (corrLocal)  atomicAdd(&sCorrect, corrLocal);
  if (nPredLocal) atomicAdd(&sNPred,   nPredLocal);
  __syncthreads();

  // ---------------- phase 5: deterministic block loss reduction + global accumulate ----------------
  if (t < 32) {
    float s = sVal[t] + sVal[t+32] + sVal[t+64];
#pragma unroll
    for (int off = 16; off > 0; off >>= 1) s += __shfl_down(s, off, 32);
    if (t == 0) {
      atomicAdd(loss_acc, (double)s);
      atomicAdd(corr_acc,  (unsigned long long)sCorrect);
      atomicAdd(npred_acc, (unsigned long long)sNPred);
    }
  }
}

__global__ void tdoa_init(unsigned long long* ws) {
  ws[0] = 0ull; ws[1] = 0ull; ws[2] = 0ull;
}

__global__ void tdoa_fin(const double* loss, const unsigned long long* cnts, float* out) {
  out[0] = (float)(loss[0] / 192000.0);
  unsigned long long c = cnts[0], n = cnts[1];
  out[1] = (n > 0ull) ? ((float)c / (float)n) : 0.0f;
}

extern "C" void kernel_launch(void* const* d_in, const int* in_sizes, int n_in,
                              void* d_out, int out_size, void* d_ws, size_t ws_size,
                              hipStream_t stream) {
  const float* pred   = (const float*)d_in[0];
  const float* target = (const float*)d_in[1];
  const float* mic    = (const float*)d_in[2];
  double* loss_acc = (double*)d_ws;
  unsigned long long* cnts = (unsigned long long*)((char*)d_ws + 8);

  tdoa_init<<<1, 1, 0, stream>>>((unsigned long long*)d_ws);
  tdoa_main<<<NBLK, NTHREADS, 0, stream>>>(pred, target, mic, loss_acc, cnts, cnts + 1);
  tdoa_fin<<<1, 1, 0, stream>>>(loss_acc, cnts, (float*)d_out);
}